// SparseSelect_37005438222839
// MI455X (gfx1250) — compile-verified
//
#include <hip/hip_runtime.h>

// ---------------------------------------------------------------------------
// Problem constants (reference: N,C,H,W = 8,256,100,152; M=32000; K=3)
// ---------------------------------------------------------------------------
#define CC 256
#define HH 100
#define WW 152

// ---------------------------------------------------------------------------
// CDNA5 async Global<->LDS path availability (confirmed present in round 2)
// ---------------------------------------------------------------------------
#if __has_builtin(__builtin_amdgcn_global_load_async_to_lds_b32) && \
    __has_builtin(__builtin_amdgcn_global_store_async_from_lds_b128)
#define HAVE_ASYNC_LDS 1
#else
#define HAVE_ASYNC_LDS 0
#endif

#if HAVE_ASYNC_LDS

// Builtin-exact pointer types:
//   load_async_to_lds_b32  : (int  AS1*, int  AS3*, imm int, imm int)
//   store_async_from_lds_b128: (v4i AS1*, v4i AS3*, imm int, imm int)
typedef __attribute__((address_space(1))) int gint_t;
typedef __attribute__((address_space(3))) int lint_t;
typedef int v4i_t __attribute__((vector_size(16)));
typedef __attribute__((address_space(1))) v4i_t gv4i_t;
typedef __attribute__((address_space(3))) v4i_t lv4i_t;

// Generic -> AS1 via ptrtoint/inttoptr (values are identical for global).
__device__ __forceinline__ gint_t* gp32(const void* p) {
    return (gint_t*)(unsigned long long)(size_t)p;
}
__device__ __forceinline__ gv4i_t* gp128(const void* p) {
    return (gv4i_t*)(unsigned long long)(size_t)p;
}
// Generic flat pointer to a __shared__ object: low 32 bits are the LDS byte
// address (flat LDS aperture = {aperture_hi32, lds_offset32}); AS3 ptrs are 32b.
__device__ __forceinline__ lint_t* lp32(void* p) {
    return (lint_t*)(unsigned)(size_t)p;
}
__device__ __forceinline__ lv4i_t* lp128(void* p) {
    return (lv4i_t*)(unsigned)(size_t)p;
}

__device__ __forceinline__ void wait_async_all() {
#if __has_builtin(__builtin_amdgcn_s_wait_asynccnt)
    __builtin_amdgcn_s_wait_asynccnt(0);
#else
    asm volatile("s_wait_asynccnt 0" ::: "memory");
#endif
}

// out[m, c, ky, kx] = features[b, c, y+ky-1, x+kx-1] (zero outside bounds).
// Async pipeline: gather -> LDS (output order) -> async b128 store to global.
__global__ __launch_bounds__(CC) void SparseSelect_gather3x3_async(
    const float* __restrict__ f,
    const int* __restrict__ batches,
    const int* __restrict__ offsets,
    float* __restrict__ out)
{
    __shared__ __align__(16) float buf[CC * 9];  // 9216 B staging, output layout

    const int m = blockIdx.x;
    const int c = threadIdx.x;

    const int b   = batches[m];
    const int off = offsets[m];
    const int y   = off / WW;
    const int x   = off - y * WW;

    const float* __restrict__ fc = f + (size_t)(b * CC + c) * (size_t)(HH * WW);
    float* lds_c = &buf[c * 9];

    const bool interior = (x >= 1) & (x <= WW - 2) & (y >= 1) & (y <= HH - 2);

    if (interior) {
        const float* p = fc + (y - 1) * WW + (x - 1);
#pragma unroll
        for (int r = 0; r < 3; ++r) {
#pragma unroll
            for (int k = 0; k < 3; ++k) {
                __builtin_amdgcn_global_load_async_to_lds_b32(
                    gp32(p + r * WW + k), lp32(lds_c + r * 3 + k), 0, 0);
            }
        }
    } else {
        // Border (block-uniform path): zero OOB slots via DS, async-load the rest.
#pragma unroll
        for (int r = 0; r < 3; ++r) {
            const int  yy  = y - 1 + r;
            const bool yok = ((unsigned)yy < (unsigned)HH);
#pragma unroll
            for (int k = 0; k < 3; ++k) {
                const int  xx = x - 1 + k;
                const bool ok = yok && ((unsigned)xx < (unsigned)WW);
                if (ok) {
                    __builtin_amdgcn_global_load_async_to_lds_b32(
                        gp32(fc + yy * WW + xx), lp32(lds_c + r * 3 + k), 0, 0);
                } else {
                    lds_c[r * 3 + k] = 0.0f;  // ds_store_b32
                }
            }
        }
        asm volatile("s_wait_dscnt 0" ::: "memory");
    }

    // Our async loads wrote LDS slots that other waves' threads will store out:
    // wait per-wave async completion, then block barrier.
    wait_async_all();
    __syncthreads();

    // Write the block's dense 9216 B output chunk: 576 x b128 chunks, 16B aligned
    // on both LDS and global sides, streamed by the async DMA path (no VGPRs).
    float* __restrict__ om = out + (size_t)m * (CC * 9);
#pragma unroll
    for (int j = threadIdx.x; j < (CC * 9) / 4; j += CC) {
        __builtin_amdgcn_global_store_async_from_lds_b128(
            gp128(om + j * 4), lp128(&buf[j * 4]), 0, 0);
    }
    // S_ENDPGM performs an implicit wait-idle, draining ASYNCcnt.
}

#else  // !HAVE_ASYNC_LDS ------------------------------------------------------

__global__ __launch_bounds__(CC) void SparseSelect_gather3x3(
    const float* __restrict__ f,
    const int* __restrict__ batches,
    const int* __restrict__ offsets,
    float* __restrict__ out)
{
    const int m = blockIdx.x;
    const int c = threadIdx.x;

    const int b   = batches[m];
    const int off = offsets[m];
    const int y   = off / WW;
    const int x   = off - y * WW;

    const float* __restrict__ fc = f + (size_t)(b * CC + c) * (size_t)(HH * WW);

    float v[9];
    const bool interior = (x >= 1) & (x <= WW - 2) & (y >= 1) & (y <= HH - 2);

    if (interior) {
        const float* __restrict__ p = fc + (y - 1) * WW + (x - 1);
#pragma unroll
        for (int r = 0; r < 3; ++r) {
            const float* __restrict__ rp = p + r * WW;
            v[r * 3 + 0] = rp[0];
            v[r * 3 + 1] = rp[1];
            v[r * 3 + 2] = rp[2];
        }
    } else {
#pragma unroll
        for (int r = 0; r < 3; ++r) {
            const int  yy  = y - 1 + r;
            const bool yok = ((unsigned)yy < (unsigned)HH);
            const int  yc  = yy < 0 ? 0 : (yy > HH - 1 ? HH - 1 : yy);
            const float* __restrict__ rp = fc + yc * WW;
#pragma unroll
            for (int k = 0; k < 3; ++k) {
                const int  xx = x - 1 + k;
                const bool ok = yok && ((unsigned)xx < (unsigned)WW);
                const int  xc = xx < 0 ? 0 : (xx > WW - 1 ? WW - 1 : xx);
                const float t = rp[xc];
                v[r * 3 + k] = ok ? t : 0.0f;
            }
        }
    }

    float* __restrict__ op = out + (size_t)m * (CC * 9) + (size_t)c * 9;
#pragma unroll
    for (int i = 0; i < 9; ++i) op[i] = v[i];
}

#endif  // HAVE_ASYNC_LDS

extern "C" void kernel_launch(void* const* d_in, const int* in_sizes, int n_in,
                              void* d_out, int out_size, void* d_ws, size_t ws_size,
                              hipStream_t stream) {
    const float* f       = (const float*)d_in[0];
    const int*   batches = (const int*)d_in[1];
    const int*   offsets = (const int*)d_in[2];
    float* out = (float*)d_out;

    const int M = in_sizes[1];  // number of query points
#if HAVE_ASYNC_LDS
    SparseSelect_gather3x3_async<<<dim3(M), dim3(CC), 0, stream>>>(f, batches, offsets, out);
#else
    SparseSelect_gather3x3<<<dim3(M), dim3(CC), 0, stream>>>(f, batches, offsets, out);
#endif
}